// SymmetricContraction_2774548873725
// MI455X (gfx1250) — compile-verified
//
#include <hip/hip_runtime.h>

// ---------------------------------------------------------------------------
// MACE SymmetricContraction (K=3, D=9, irreps 0e+1o) on gfx1250.
// Reformulated as: out[b,c,i] = gather_e( X[b,:] @ Q_c )  with
//   X[b,m]   = 219 symmetric monomials of x[b,c,0..8]   (9 + 45 + 165)
//   Q_c[m,n] = species-conditioned coefficient matrix, n = e*4 + i, padded
//              to N=48 (3 WMMA N-tiles), K padded to 220 (55 chunks of 4).
// Main kernel: one workgroup per channel c, fp32 GEMM on
// v_wmma_f32_16x16x4_f32; Q_c staged into LDS by the Tensor Data Mover
// (tensor_load_to_lds + s_wait_tensorcnt), X monomials built in LDS.
// ---------------------------------------------------------------------------

typedef float v2f __attribute__((ext_vector_type(2)));
typedef float v8f __attribute__((ext_vector_type(8)));
typedef int   s4i __attribute__((ext_vector_type(4)));
typedef int   s8i __attribute__((ext_vector_type(8)));

#define B_N   512
#define C_N   128
#define D_N   9
#define E_N   10
#define NM1   9
#define NM2   45
#define NM3   165
#define M_TOT 219   // 9 + 45 + 165
#define K_PAD 220   // multiple of 4
#define N_OUT 40    // E_N * 4 outputs (0e:1 + 1o:3)
#define N_PAD 48    // 3 tiles of 16
#define QELEMS (N_PAD * K_PAD)   // 10560 f32 per channel

// ---------------------------------------------------------------------------
// species[b] = argmax_e y[b,e]   (y is one-hot)
// ---------------------------------------------------------------------------
__global__ void sc_species_kernel(const float* __restrict__ y,
                                  int* __restrict__ sp) {
  int b = blockIdx.x * blockDim.x + threadIdx.x;
  if (b >= B_N) return;
  int e = 0;
  float best = -1.0e30f;
  for (int j = 0; j < E_N; ++j) {
    float v = y[b * E_N + j];
    if (v > best) { best = v; e = j; }
  }
  sp[b] = e;
}

// ---------------------------------------------------------------------------
// Build Q[c][n][m] with n = e*4+i (i=0 -> 0e, i=1..3 -> 1o components),
// m = monomial index (order-1, then order-2 j1<=j2, then order-3 j1<=j2<=j3),
// multiplicity folded in. Padded entries (n>=40, m>=219) are zero.
// Q layout: Q[(c*N_PAD + n)*K_PAD + m]
// ---------------------------------------------------------------------------
__global__ void sc_build_q_kernel(
    const float* __restrict__ u1_0e, const float* __restrict__ w1_0e,
    const float* __restrict__ u2_0e, const float* __restrict__ w2_0e,
    const float* __restrict__ u3_0e, const float* __restrict__ w3_0e,
    const float* __restrict__ u1_1o, const float* __restrict__ w1_1o,
    const float* __restrict__ u2_1o, const float* __restrict__ w2_1o,
    const float* __restrict__ u3_1o, const float* __restrict__ w3_1o,
    float* __restrict__ Q) {
  const int total = C_N * N_PAD * K_PAD;
  int idx = blockIdx.x * blockDim.x + threadIdx.x;
  if (idx >= total) return;

  int m = idx % K_PAD;
  int n = (idx / K_PAD) % N_PAD;
  int c = idx / (K_PAD * N_PAD);

  float val = 0.0f;
  if (n < N_OUT && m < M_TOT) {
    int e = n >> 2;
    int i = n & 3;

    // decode m -> order, (j1<=j2<=j3), multiplicity
    int order, j1 = 0, j2 = 0, j3 = 0, mult = 1;
    if (m < NM1) {
      order = 1; j1 = m; mult = 1;
    } else if (m < NM1 + NM2) {
      order = 2;
      int r = m - NM1;
      j1 = 0;
      while (r >= (D_N - j1)) { r -= (D_N - j1); ++j1; }
      j2 = j1 + r;
      mult = (j1 == j2) ? 1 : 2;
    } else {
      order = 3;
      int r = m - (NM1 + NM2);
      j1 = 0;
      for (;;) {
        int nn = D_N - j1;
        int t = nn * (nn + 1) / 2;
        if (r < t) break;
        r -= t; ++j1;
      }
      j2 = j1;
      while (r >= (D_N - j2)) { r -= (D_N - j2); ++j2; }
      j3 = j2 + r;
      mult = (j1 == j2 && j2 == j3) ? 1 : ((j1 == j2 || j2 == j3) ? 3 : 6);
    }

    if (i == 0) {
      // 0e output, irdim = 1, MULS = {1,3,8}
      if (order == 1) {
        val = u1_0e[j1] * w1_0e[e * C_N + c];
      } else if (order == 2) {
        int base = (j1 * D_N + j2) * 3;
        for (int k = 0; k < 3; ++k)
          val += u2_0e[base + k] * w2_0e[(k * E_N + e) * C_N + c];
      } else {
        int base = ((j1 * D_N + j2) * D_N + j3) * 8;
        for (int k = 0; k < 8; ++k)
          val += u3_0e[base + k] * w3_0e[(k * E_N + e) * C_N + c];
      }
    } else {
      // 1o output, irdim = 3, MULS = {1,4,12}
      int ii = i - 1;
      if (order == 1) {
        val = u1_1o[j1 * 3 + ii] * w1_1o[e * C_N + c];
      } else if (order == 2) {
        int base = (j1 * D_N + j2) * 4 * 3;
        for (int k = 0; k < 4; ++k)
          val += u2_1o[base + k * 3 + ii] * w2_1o[(k * E_N + e) * C_N + c];
      } else {
        int base = ((j1 * D_N + j2) * D_N + j3) * 12 * 3;
        for (int k = 0; k < 12; ++k)
          val += u3_1o[base + k * 3 + ii] * w3_1o[(k * E_N + e) * C_N + c];
      }
    }
    val *= (float)mult;
  }
  Q[idx] = val;
}

// ---------------------------------------------------------------------------
// Main kernel: one workgroup (256 threads = 8 waves) per channel c.
//   0) TDM: tensor_load_to_lds DMAs the 42KB Q_c block Global->LDS (wave 0)
//   per pass of 64 batch rows:
//   1) build 219 monomials per row into LDS (4 threads per row)
//   2) 12 WMMA tile-jobs (4 M-tiles x 3 N-tiles), K = 55 chunks of 4,
//      v_wmma_f32_16x16x4_f32 with A from Xs, B from Qs (K-contiguous)
//   3) gather out[b,c,i] = Y[b, e(b)*4 + i]
// ---------------------------------------------------------------------------
__launch_bounds__(256, 1)
__global__ void sc_wmma_kernel(const float* __restrict__ x,
                               const float* __restrict__ Q,
                               const int* __restrict__ sp,
                               float* __restrict__ out) {
  __shared__ float Qs[N_PAD][K_PAD];   // 42,240 B  (stored [n][k])
  __shared__ float Xs[64][K_PAD];      // 56,320 B  (stored [row][k])
  __shared__ float Ys[64][N_PAD];      // 12,288 B

  const int c    = blockIdx.x;
  const int tid  = threadIdx.x;
  const int lane = tid & 31;
  const int wave = tid >> 5;

  // ---- 0) stage Q_c into LDS via the Tensor Data Mover ----
  // D# per CDNA5 ISA: 2 SGPR groups (<=2D tensor).
  // 1-row tensor: tensor_dim0 = tile_dim0 = QELEMS f32, data_size = 4B.
  const float* Qc = Q + (size_t)c * QELEMS;
  if (wave == 0) {
    const unsigned long long ga = (unsigned long long)(uintptr_t)Qc;
    const unsigned int lds_off  = (unsigned int)(uintptr_t)(&Qs[0][0]);

    s4i g0;
    g0.x = 1;                                   // count=1 (valid user D#)
    g0.y = (int)lds_off;                        // lds_addr [63:32]
    g0.z = (int)(unsigned int)ga;               // global_addr[31:0]
    g0.w = (int)(0x80000000u |                  // type=2 at [127:126]
                 ((unsigned int)(ga >> 32) & 0x01FFFFFFu)); // global_addr[56:32]

    s8i g1;
    g1.s0 = 0x00020000;                         // data_size=2 (4B); wg_mask=0
    g1.s1 = (int)((QELEMS & 0xFFFFu) << 16);    // tensor_dim0[15:0] at [63:48]
    g1.s2 = (int)(((QELEMS >> 16) & 0xFFFFu)    // tensor_dim0[31:16]
                  | (1u << 16));                // tensor_dim1 = 1
    g1.s3 = (int)((QELEMS & 0xFFFFu) << 16);    // tile_dim0 = QELEMS
    g1.s4 = 1;                                  // tile_dim1 = 1, tile_dim2 = 0
    g1.s5 = (int)QELEMS;                        // tensor_dim0_stride[31:0]
    g1.s6 = 0;                                  // stride hi / dim1_stride lo
    g1.s7 = 0;

    asm volatile("tensor_load_to_lds %0, %1" :: "s"(g0), "s"(g1) : "memory");
    __builtin_amdgcn_s_wait_tensorcnt(0);
  }
  __syncthreads();

  for (int pass = 0; pass < 8; ++pass) {
    const int b0 = pass * 64;

    // ---- 1) build monomials: 4 threads per batch row ----
    {
      const int row  = tid >> 2;     // 0..63
      const int part = tid & 3;
      const int b    = b0 + row;
      const float* xr = x + ((size_t)b * C_N + c) * D_N;
      if (pass < 7) {
        // speculative prefetch of next pass's x rows (global_prefetch_b8)
        __builtin_prefetch(xr + (size_t)64 * C_N * D_N, 0, 0);
      }
      float xv[D_N];
#pragma unroll
      for (int j = 0; j < D_N; ++j) xv[j] = xr[j];

      int m = 0;
#pragma unroll
      for (int ja = 0; ja < D_N; ++ja) {
        if ((m & 3) == part) Xs[row][m] = xv[ja];
        ++m;
      }
#pragma unroll
      for (int ja = 0; ja < D_N; ++ja)
#pragma unroll
        for (int jb = ja; jb < D_N; ++jb) {
          if ((m & 3) == part) Xs[row][m] = xv[ja] * xv[jb];
          ++m;
        }
      for (int ja = 0; ja < D_N; ++ja)
        for (int jb = ja; jb < D_N; ++jb) {
          float p = xv[ja] * xv[jb];
          for (int jc = jb; jc < D_N; ++jc) {
            if ((m & 3) == part) Xs[row][m] = p * xv[jc];
            ++m;
          }
        }
      if (part == 3) Xs[row][M_TOT] = 0.0f;  // K padding
    }
    __syncthreads();

    // ---- 2) WMMA GEMM: (64 x 220) @ (220 x 48) ----
    // job = nt*4 + mt ; waves 0..7 take jobs {w, w+8}
    for (int job = wave; job < 12; job += 8) {
      const int mt = job & 3;
      const int nt = job >> 2;

      v8f acc = {0.f, 0.f, 0.f, 0.f, 0.f, 0.f, 0.f, 0.f};

      // A frag: lanes 0-15 -> M rows, VGPR0/1 = K0/K1; lanes 16-31 -> K2/K3
      const int rA  = mt * 16 + (lane & 15);
      const int nB  = nt * 16 + (lane & 15);
      const int khi = (lane >> 4) * 2;

      for (int kc = 0; kc < K_PAD / 4; ++kc) {
        const int kk = kc * 4 + khi;
        v2f a  = *(const v2f*)&Xs[rA][kk];
        v2f bb = *(const v2f*)&Qs[nB][kk];
        acc = __builtin_amdgcn_wmma_f32_16x16x4_f32(
            /*neg_a=*/false, a, /*neg_b=*/false, bb,
            /*c_mod=*/(short)0, acc, /*reuse_a=*/false, /*reuse_b=*/false);
      }

      // D layout: VGPR r -> M = r + 8*(lane>=16), N = lane%16
      const int hi = lane >> 4;
      const int nW = nt * 16 + (lane & 15);
#pragma unroll
      for (int r = 0; r < 8; ++r)
        Ys[mt * 16 + r + 8 * hi][nW] = acc[r];
    }
    __syncthreads();

    // ---- 3) gather by species and store ----
    {
      const int row = tid >> 2;
      const int i   = tid & 3;
      const int b   = b0 + row;
      const int e   = sp[b];
      out[((size_t)b * C_N + c) * 4 + i] = Ys[row][e * 4 + i];
    }
    __syncthreads();
  }
}

// ---------------------------------------------------------------------------
extern "C" void kernel_launch(void* const* d_in, const int* in_sizes, int n_in,
                              void* d_out, int out_size, void* d_ws, size_t ws_size,
                              hipStream_t stream) {
  (void)in_sizes; (void)n_in; (void)out_size; (void)ws_size;

  const float* x     = (const float*)d_in[0];
  const float* y     = (const float*)d_in[1];
  const float* u1_0e = (const float*)d_in[2];
  const float* w1_0e = (const float*)d_in[3];
  const float* u2_0e = (const float*)d_in[4];
  const float* w2_0e = (const float*)d_in[5];
  const float* u3_0e = (const float*)d_in[6];
  const float* w3_0e = (const float*)d_in[7];
  const float* u1_1o = (const float*)d_in[8];
  const float* w1_1o = (const float*)d_in[9];
  const float* u2_1o = (const float*)d_in[10];
  const float* w2_1o = (const float*)d_in[11];
  const float* u3_1o = (const float*)d_in[12];
  const float* w3_1o = (const float*)d_in[13];

  // workspace: Q (128*48*220 f32 = 5,406,720 B) then species (512 i32)
  float* Q = (float*)d_ws;
  int*   sp = (int*)((char*)d_ws + (size_t)C_N * N_PAD * K_PAD * sizeof(float));

  sc_species_kernel<<<(B_N + 255) / 256, 256, 0, stream>>>(y, sp);

  const int q_total = C_N * N_PAD * K_PAD;
  sc_build_q_kernel<<<(q_total + 255) / 256, 256, 0, stream>>>(
      u1_0e, w1_0e, u2_0e, w2_0e, u3_0e, w3_0e,
      u1_1o, w1_1o, u2_1o, w2_1o, u3_1o, w3_1o, Q);

  sc_wmma_kernel<<<C_N, 256, 0, stream>>>(x, Q, sp, (float*)d_out);
}